// RFCN_47699906789424
// MI455X (gfx1250) — compile-verified
//
#include <hip/hip_runtime.h>
#include <hip/hip_bf16.h>
#include <math.h>

// ---------------- problem constants (from reference) ----------------
#define CIN      1024
#define Hh       64
#define Ww       64
#define HW       4096      // 64*64
#define BATCH    2
#define NPROP    1000
#define NCLS     21        // NUM_CLASSES+1
#define NREG     4
#define NGROUPS  25        // 21 + 4 effective channels after K*K fold
#define MPAD     32        // padded M for WMMA tiles
#define KK       9         // K*K
#define KSPLIT   4         // K-dimension split (chunks of 256)
#define KCHUNK   (CIN / KSPLIT)
#define NTILE    32        // columns per wave
#define NTILES   (HW / NTILE)   // 128 per batch

typedef __attribute__((ext_vector_type(2))) float v2f;
typedef __attribute__((ext_vector_type(8))) float v8f;

// -------------------------------------------------------------------
// Kernel 1: fold the K*K channel-group sum into the weights.
// w_eff[g][c] = sum_j w[g*9+j][c]  (g<21 from w_cls, 21<=g<25 from w_reg)
// rows 25..31 zero-padded so the WMMA M=32 tile is clean.
// -------------------------------------------------------------------
__global__ void reduce_w_kernel(const float* __restrict__ wcls,
                                const float* __restrict__ bcls,
                                const float* __restrict__ wreg,
                                const float* __restrict__ breg,
                                float* __restrict__ weff,
                                float* __restrict__ beff) {
    const int g = blockIdx.x;                 // 0..31
    for (int c = threadIdx.x; c < CIN; c += blockDim.x) {
        float s = 0.f;
        if (g < NCLS) {
            #pragma unroll
            for (int j = 0; j < KK; ++j) s += wcls[(g * KK + j) * CIN + c];
        } else if (g < NGROUPS) {
            const int gr = g - NCLS;
            #pragma unroll
            for (int j = 0; j < KK; ++j) s += wreg[(gr * KK + j) * CIN + c];
        }
        weff[g * CIN + c] = s;                // zero for padded rows
    }
    if (threadIdx.x == 0) {
        float s = 0.f;
        if (g < NCLS) {
            #pragma unroll
            for (int j = 0; j < KK; ++j) s += bcls[g * KK + j];
        } else if (g < NGROUPS) {
            const int gr = g - NCLS;
            #pragma unroll
            for (int j = 0; j < KK; ++j) s += breg[gr * KK + j];
        }
        beff[g] = s;
    }
}

// -------------------------------------------------------------------
// Kernel 2: fp32 WMMA GEMM partials.
//   part[q][b][g][hw] = w_eff[g, q*256:(q+1)*256] . feat[b, q*256:(q+1)*256, hw]
// One wave per (batch, 32-column tile, k-chunk): 2*128*4 = 1024 waves.
// M=32 (2 x 16-row WMMA tiles), N=32 (2 x 16-col tiles), K=256 per chunk,
// stepped 4 at a time with V_WMMA_F32_16X16X4_F32 (bit-exact fp32).
//
// A-matrix 16x4 f32 layout (ISA 7.12.2): lanes 0-15 hold M=lane, VGPR0=K0,
// VGPR1=K1; lanes 16-31 hold M=lane-16, VGPR0=K2, VGPR1=K3.
// B-matrix 4x16: lane<16 -> col N=lane rows K0,K1; lane>=16 -> rows K2,K3.
// D 16x16 f32: VGPR r, lane<16 -> M=r; lane>=16 -> M=r+8.
// -------------------------------------------------------------------
__global__ void gemm_wmma_kernel(const float* __restrict__ feat,
                                 const float* __restrict__ weff,
                                 float* __restrict__ part) {
    const int wavesPerBlock = blockDim.x >> 5;
    const int waveId = blockIdx.x * wavesPerBlock + (threadIdx.x >> 5);
    const int lane   = threadIdx.x & 31;

    const int b      = waveId >> 9;            // 512 waves per batch
    const int within = waveId & 511;
    const int ntile  = within >> 2;            // 0..127
    const int q      = within & 3;             // k-chunk
    const int n0     = ntile * NTILE;
    const int kbase  = q * KCHUNK;

    const int lrow  = lane & 15;
    const int khalf = lane >> 4;               // 0: K0/K1   1: K2/K3

    const float* __restrict__ fb = feat + (size_t)b * CIN * HW;

    v8f acc[2][2] = {};                        // [m-tile][n-tile]

    for (int k0 = kbase; k0 < kbase + KCHUNK; k0 += 4) {
        const int k = k0 + khalf * 2;
        // A fragments (w_eff rows): contiguous pair -> b64 load, L2-resident
        v2f a0, a1;
        {
            const float* ap0 = weff + (0 * 16 + lrow) * CIN + k;
            const float* ap1 = weff + (1 * 16 + lrow) * CIN + k;
            a0.x = ap0[0]; a0.y = ap0[1];
            a1.x = ap1[0]; a1.y = ap1[1];
        }
        #pragma unroll
        for (int j = 0; j < 2; ++j) {
            const int n = n0 + j * 16 + lrow;
            const float* bp = fb + (size_t)k * HW + n;
            v2f bb;
            bb.x = bp[0];
            bb.y = bp[HW];
            acc[0][j] = __builtin_amdgcn_wmma_f32_16x16x4_f32(
                false, a0, false, bb, (short)0, acc[0][j], false, false);
            acc[1][j] = __builtin_amdgcn_wmma_f32_16x16x4_f32(
                false, a1, false, bb, (short)0, acc[1][j], false, false);
        }
    }

    // store 25 real rows of this partial
    float* __restrict__ pq = part + ((size_t)q * BATCH + b) * NGROUPS * HW;
    const int mofs = (lane >= 16) ? 8 : 0;
    #pragma unroll
    for (int t = 0; t < 2; ++t) {
        #pragma unroll
        for (int j = 0; j < 2; ++j) {
            const int n = n0 + j * 16 + lrow;
            #pragma unroll
            for (int r = 0; r < 8; ++r) {
                const int m = t * 16 + r + mofs;
                if (m < NGROUPS) {
                    pq[(size_t)m * HW + n] = acc[t][j][r];
                }
            }
        }
    }
}

// -------------------------------------------------------------------
// Kernel 3: deterministic reduction of the 4 K-chunk partials + bias.
// scores[b][g][hw] = sum_q part[q][b][g][hw] + b_eff[g]
// -------------------------------------------------------------------
__global__ void reduce_scores_kernel(const float* __restrict__ part,
                                     const float* __restrict__ beff,
                                     float* __restrict__ scores) {
    const size_t i = (size_t)blockIdx.x * blockDim.x + threadIdx.x;
    const size_t total = (size_t)BATCH * NGROUPS * HW;    // 204800
    if (i < total) {
        const int g = (int)((i / HW) % NGROUPS);
        float s = beff[g];
        #pragma unroll
        for (int qq = 0; qq < KSPLIT; ++qq)
            s += part[(size_t)qq * total + i];            // fixed order -> deterministic
        scores[i] = s;
    }
}

// -------------------------------------------------------------------
// Kernel 4: PS-ROI bin(0,0) average per proposal. One wave per proposal,
// lane = effective channel (25 used). Scores are L2-resident (0.8 MB).
// -------------------------------------------------------------------
__global__ void psroi_kernel(const float* __restrict__ scores,
                             const int* __restrict__ props,
                             float* __restrict__ out) {
    const int idx = blockIdx.x;                // 0..B*NPROP-1
    const int b = idx / NPROP;
    const int n = idx - b * NPROP;
    const int g = threadIdx.x;                 // lane

    const int* p = props + ((size_t)b * NPROP + n) * 4;
    const float inv = 1.0f / 32.0f;            // STRIDE=32, exact
    const int x1 = (int)floorf((float)p[0] * inv);
    const int y1 = (int)floorf((float)p[1] * inv);
    const int x2 = (int)ceilf ((float)p[2] * inv);
    const int y2 = (int)ceilf ((float)p[3] * inv);
    const int hb = (y2 - y1 + 2) / 3;          // ceil(h/K), K=3
    const int wb = (x2 - x1 + 2) / 3;
    const int r0 = max(y1, 0), r1 = min(y1 + hb, Hh);
    const int c0 = max(x1, 0), c1 = min(x1 + wb, Ww);
    const float denom = (float)(hb * wb);

    if (g < NGROUPS) {
        const float* sp = scores + ((size_t)b * NGROUPS + g) * HW;
        float s = 0.f;
        for (int r = r0; r < r1; ++r) {
            const float* row = sp + r * Ww;
            for (int c = c0; c < c1; ++c) s += row[c];
        }
        s /= denom;
        if (g < NCLS) {
            out[((size_t)b * NPROP + n) * NCLS + g] = s;                       // cls block
        } else {
            out[(size_t)BATCH * NPROP * NCLS +
                ((size_t)b * NPROP + n) * NREG + (g - NCLS)] = s;              // reg block
        }
    }
}

// -------------------------------------------------------------------
extern "C" void kernel_launch(void* const* d_in, const int* in_sizes, int n_in,
                              void* d_out, int out_size, void* d_ws, size_t ws_size,
                              hipStream_t stream) {
    const float* feat = (const float*)d_in[0];   // [2,1024,64,64]
    const float* wcls = (const float*)d_in[1];   // [189,1024]
    const float* bcls = (const float*)d_in[2];   // [189]
    const float* wreg = (const float*)d_in[3];   // [36,1024]
    const float* breg = (const float*)d_in[4];   // [36]
    const int*   prop = (const int*)  d_in[5];   // [2,1000,4]
    float* out = (float*)d_out;                  // 42000 cls + 8000 reg

    char* ws = (char*)d_ws;
    const size_t weff_bytes  = (size_t)MPAD * CIN * 4;                 // 128 KB
    const size_t part_bytes  = (size_t)KSPLIT * BATCH * NGROUPS * HW * 4; // 3.2 MB
    float* weff   = (float*)ws;
    float* beff   = (float*)(ws + weff_bytes);                          // 128 B
    float* part   = (float*)(ws + weff_bytes + 256);
    float* scores = (float*)(ws + weff_bytes + 256 + part_bytes);       // 800 KB

    // 1) fold K*K group-sum into weights/bias (32 rows, zero-padded)
    reduce_w_kernel<<<MPAD, 256, 0, stream>>>(wcls, bcls, wreg, breg, weff, beff);

    // 2) fp32 WMMA GEMM partials: 1024 waves = 128 blocks x 8 waves
    gemm_wmma_kernel<<<128, 256, 0, stream>>>(feat, weff, part);

    // 3) deterministic K-chunk reduction + bias: 204800 elems
    reduce_scores_kernel<<<(BATCH * NGROUPS * HW + 255) / 256, 256, 0, stream>>>(
        part, beff, scores);

    // 4) per-proposal pooled outputs
    psroi_kernel<<<BATCH * NPROP, 32, 0, stream>>>(scores, prop, out);
}